// ChamferDistance_73160472920067
// MI455X (gfx1250) — compile-verified
//
#include <hip/hip_runtime.h>

typedef float v2f __attribute__((ext_vector_type(2)));
typedef float v8f __attribute__((ext_vector_type(8)));

#define WAVES_PER_BLOCK 8

// D = A(16x4) * B(4x16) + 0 via V_WMMA_F32_16X16X4_F32.
// A row m = [px, py, pz, 1];  B col n = [qx, qy, qz, -0.5*|q|^2]
//   => C[m][n] = p.q - 0.5|q|^2, and min_n dist^2 = max(|p|^2 - 2*max_n C, 0).
// One wave owns a 16-row strip of P and scans all of Q; inner loop is
// 1x b96 load + ~6 VALU + 1 WMMA + 4 v_max3 per 16-column tile.
__global__ void __launch_bounds__(256)
chamfer_dual_kernel(const float* __restrict__ p1, const float* __restrict__ p2,
                    float* __restrict__ part12, float* __restrict__ part21,
                    int N, int M, int stripsN, int stripsM) {
  const int lane = threadIdx.x;   // 0..31 (wave32)
  const int wave = threadIdx.y;   // 0..WAVES_PER_BLOCK-1
  const int b    = blockIdx.y;

  const float *P, *Q; float* partial; int Np, Mq, strips;
  if (blockIdx.z == 0) { P = p1; Q = p2; partial = part12; Np = N; Mq = M; strips = stripsN; }
  else                 { P = p2; Q = p1; partial = part21; Np = M; Mq = N; strips = stripsM; }

  const int strip = blockIdx.x * WAVES_PER_BLOCK + wave;
  if (strip >= strips) return;    // uniform per wave

  const float* Pb = P + (size_t)b * Np * 3;
  const float* Qb = Q + (size_t)b * Mq * 3;

  const bool hi = lane >= 16;
  const int  li = lane & 15;

  // ---- A fragment: lanes 0-15 hold K=0,1 (x,y); lanes 16-31 hold K=2,3 (z,1).
  int r = strip * 16 + li;
  if (r >= Np) r = Np - 1;        // address safety (Np is a multiple of 16 here)
  const float* prow = Pb + (size_t)r * 3;
  float px = prow[0], py = prow[1], pz = prow[2];
  float asq = px * px + py * py + pz * pz;   // |P[r]|^2 (same in lane l and l^16)
  v2f A; A[0] = hi ? pz : px; A[1] = hi ? 1.0f : py;

  float rowmax[8];
#pragma unroll
  for (int j = 0; j < 8; ++j) rowmax[j] = -3.0e38f;

  // Running per-lane pointer: unrolled loads become base + IOFFSET immediates.
  const int nTiles = Mq >> 4;
  const float* qptr = Qb + (size_t)li * 3;
#pragma unroll 4
  for (int t = 0; t < nTiles; ++t) {
    float qx = qptr[0], qy = qptr[1], qz = qptr[2];   // one b96 per lane
    qptr += 48;                                       // 16 points * 3 floats
    float qsq = qx * qx + qy * qy + qz * qz;
    v2f Bv; Bv[0] = hi ? qz : qx; Bv[1] = hi ? (-0.5f * qsq) : qy;
    v8f C = {};
    C = __builtin_amdgcn_wmma_f32_16x16x4_f32(false, A, false, Bv, (short)0, C,
                                              false, false);
#pragma unroll
    for (int j = 0; j < 8; ++j) rowmax[j] = fmaxf(rowmax[j], C[j]);
  }

  const int MqFull = nTiles << 4;
  if (MqFull < Mq) {              // tail tile: poison invalid columns' K=3 slot
    const int  c     = MqFull + li;
    const bool valid = c < Mq;
    const float* qrow = Qb + (size_t)(valid ? c : 0) * 3;
    float qx = qrow[0], qy = qrow[1], qz = qrow[2];
    float qsq  = qx * qx + qy * qy + qz * qz;
    float b1hi = valid ? (-0.5f * qsq) : -3.0e38f;    // C -> -huge, never wins max
    v2f Bv; Bv[0] = hi ? qz : qx; Bv[1] = hi ? b1hi : qy;
    v8f C = {};
    C = __builtin_amdgcn_wmma_f32_16x16x4_f32(false, A, false, Bv, (short)0, C,
                                              false, false);
#pragma unroll
    for (int j = 0; j < 8; ++j) rowmax[j] = fmaxf(rowmax[j], C[j]);
  }

  // ---- Epilogue: fold the 16 column-lanes (xor net stays inside each half),
  // apply |p|^2, clamp, and sum the 16 per-row minima of the strip.
  float s = 0.0f;
#pragma unroll
  for (int j = 0; j < 8; ++j) {
    float v = rowmax[j];
    v = fmaxf(v, __shfl_xor(v, 1, 32));
    v = fmaxf(v, __shfl_xor(v, 2, 32));
    v = fmaxf(v, __shfl_xor(v, 4, 32));
    v = fmaxf(v, __shfl_xor(v, 8, 32));
    float rsq = __shfl(asq, j + (hi ? 8 : 0), 32);    // |P| for row M=j+8*hi
    s += fmaxf(rsq - 2.0f * v, 0.0f);                  // min_n dist^2 for that row
  }
  float s_hi = __shfl(s, 16, 32);
  if (lane == 0) partial[(size_t)b * strips + strip] = s + s_hi;
}

// Deterministic single-block reduction: fixed strided per-thread sums + LDS tree.
__global__ void __launch_bounds__(256)
chamfer_finalize_kernel(const float* __restrict__ p12, int n12,
                        const float* __restrict__ p21, int n21,
                        float inv12, float inv21, float* __restrict__ out) {
  __shared__ float sm[256];
  const int t = threadIdx.x;
  float s12 = 0.0f, s21 = 0.0f;
  for (int i = t; i < n12; i += 256) s12 += p12[i];
  for (int i = t; i < n21; i += 256) s21 += p21[i];
  sm[t] = s12 * inv12 + s21 * inv21;
  __syncthreads();
  for (int k = 128; k > 0; k >>= 1) {
    if (t < k) sm[t] += sm[t + k];
    __syncthreads();
  }
  if (t == 0) out[0] = sm[0];
}

extern "C" void kernel_launch(void* const* d_in, const int* in_sizes, int n_in,
                              void* d_out, int out_size, void* d_ws, size_t ws_size,
                              hipStream_t stream) {
  const float* p1 = (const float*)d_in[0];
  const float* p2 = (const float*)d_in[1];
  float* out = (float*)d_out;

  const int B = 4, D = 3;
  const int N = in_sizes[0] / (B * D);
  const int M = in_sizes[1] / (B * D);
  const int stripsN = (N + 15) / 16;
  const int stripsM = (M + 15) / 16;

  float* partial12 = (float*)d_ws;                       // B*stripsN floats
  float* partial21 = partial12 + (size_t)B * stripsN;    // B*stripsM floats

  const int gxN = (stripsN + WAVES_PER_BLOCK - 1) / WAVES_PER_BLOCK;
  const int gxM = (stripsM + WAVES_PER_BLOCK - 1) / WAVES_PER_BLOCK;
  dim3 blk(32, WAVES_PER_BLOCK);
  dim3 grid((gxN > gxM ? gxN : gxM), B, 2);   // z=0: d12, z=1: d21 (one launch)

  chamfer_dual_kernel<<<grid, blk, 0, stream>>>(p1, p2, partial12, partial21,
                                                N, M, stripsN, stripsM);
  chamfer_finalize_kernel<<<1, 256, 0, stream>>>(partial12, B * stripsN,
                                                 partial21, B * stripsM,
                                                 1.0f / ((float)B * (float)N),
                                                 1.0f / ((float)B * (float)M), out);
}